// MultiMessagePassingWithAttention_56418690400743
// MI455X (gfx1250) — compile-verified
//
#include <hip/hip_runtime.h>
#include <math.h>

// ---------------------------------------------------------------------------
// MultiMessagePassingWithAttention for MI455X (gfx1250, wave32, fp32 WMMA)
//
// N=4096 nodes (64 graphs x 64), D=128, HEADS=3, STEPS=4, E=65536 sparse
// edges (1024/graph, contiguous). ~8 GFLOP fp32 total, ~12 MB live data ->
// fully L2-resident: latency/launch bound, so keep native fp32 precision and
// use V_WMMA_F32_16X16X4_F32 for every matmul, fuse per-graph work into
// single LDS-resident workgroups (320 KB LDS per WGP on CDNA5).
//
// Per step (3 kernels):
//   1) msg_segmax : per graph, WMMA xm=leaky(x@Wm+bm) into LDS, then
//                   LDS-atomic segment-max over the graph's 1024 edges.
//   2) gemm_z     : x = leaky([x | x_att[batch] | agg] @ Wa + ba) + x  (WMMA)
//   3) gat_fused  : per graph/head, WMMA sl/sr projections into LDS,
//                   VALU logits (leaky inside sum - non-factorable),
//                   softmax, WMMA alpha@sl accumulated across heads.
// ---------------------------------------------------------------------------

typedef __attribute__((ext_vector_type(2))) float v2f;
typedef __attribute__((ext_vector_type(8))) float v8f;

#define NNODES 4096
#define DIM    128
#define NGRAPH 64
#define NPER   64
#define NEDGE  65536
#define EPG    1024   // edges per graph

__device__ __forceinline__ float leaky01(float v) { return v < 0.f ? 0.01f * v : v; }

__device__ __forceinline__ v8f wmma_step(v2f a, v2f b, v8f c) {
  // D = A(16x4,f32) * B(4x16,f32) + C(16x16,f32)
  return __builtin_amdgcn_wmma_f32_16x16x4_f32(false, a, false, b,
                                               (short)0, c, false, false);
}

// A fragment from row-major matrix: a[v] = A[m][k + 2*hi + v]
__device__ __forceinline__ v2f afrag(const float* A, int lda, int m, int k2) {
  v2f a;
  a.x = A[m * lda + k2];
  a.y = A[m * lda + k2 + 1];
  return a;
}
// B fragment from row-major matrix: b[v] = B[k + 2*hi + v][n]
__device__ __forceinline__ v2f bfrag(const float* B, int ldb, int k2, int col) {
  v2f b;
  b.x = B[(size_t)k2 * ldb + col];
  b.y = B[(size_t)(k2 + 1) * ldb + col];
  return b;
}

// ---------------------------------------------------------------------------
// 1) Fused edge-MLP + segment-max, one workgroup per graph.
//    leaky(x[src]@Wm+bm) == gather(leaky(x@Wm+bm)) so compute the 64x128
//    message matrix with WMMA straight into LDS and max-reduce the edges
//    there (order-preserving uint encoding + ds atomic max; sentinel 0 =
//    "no incoming edge" -> 0.0 like the reference).
// ---------------------------------------------------------------------------
__global__ __launch_bounds__(256) void msg_segmax_kernel(
    const float* __restrict__ X, const float* __restrict__ Wm,
    const float* __restrict__ bm, const int* __restrict__ src,
    const int* __restrict__ dst, float* __restrict__ AGG) {
  __shared__ float    xg[NPER * DIM];   // 32 KB node features
  __shared__ float    xm[NPER * DIM];   // 32 KB messages
  __shared__ unsigned am[NPER * DIM];   // 32 KB encoded running max
  __shared__ int es[EPG];
  __shared__ int ed[EPG];
  const int g = blockIdx.x, tid = threadIdx.x;
  const int nbase = g * NPER, ebase = g * EPG;

  for (int t = tid; t < NPER * DIM; t += 256) {
    xg[t] = X[(size_t)nbase * DIM + t];
    am[t] = 0u;
  }
  for (int t = tid; t < EPG; t += 256) {
    es[t] = src[ebase + t] - nbase;
    ed[t] = dst[ebase + t] - nbase;
  }
  __syncthreads();

  // 64x128 = (4 row tiles) x (8 col tiles); 8 waves own 4 C tiles each.
  const int wv = tid >> 5, lane = tid & 31;
  const int hi = lane >> 4, mn = lane & 15;
  const int rt = wv >> 1;         // row tile 0..3
  const int cg = (wv & 1) * 4;    // col tiles cg..cg+3
  v8f acc[4] = {};
  for (int k = 0; k < DIM; k += 4) {
    const int k2 = k + 2 * hi;
    const v2f a = afrag(xg, DIM, rt * 16 + mn, k2);
#pragma unroll
    for (int t = 0; t < 4; ++t) {
      acc[t] = wmma_step(a, bfrag(Wm, DIM, k2, (cg + t) * 16 + mn), acc[t]);
    }
  }
#pragma unroll
  for (int t = 0; t < 4; ++t) {
    const int col = (cg + t) * 16 + mn;
    const float bv = bm[col];
#pragma unroll
    for (int r = 0; r < 8; ++r)
      xm[(rt * 16 + r + 8 * hi) * DIM + col] = leaky01(acc[t][r] + bv);
  }
  __syncthreads();

  for (int w = tid; w < EPG * DIM; w += 256) {
    const int e = w >> 7, c = w & 127;
    const float v = xm[es[e] * DIM + c];
    const unsigned u = __float_as_uint(v);
    const unsigned enc = (u & 0x80000000u) ? ~u : (u | 0x80000000u);
    atomicMax(&am[ed[e] * DIM + c], enc);
  }
  __syncthreads();

  for (int t = tid; t < NPER * DIM; t += 256) {
    const unsigned e = am[t];
    const float v = (e == 0u) ? 0.0f
                  : ((e & 0x80000000u) ? __uint_as_float(e & 0x7fffffffu)
                                       : __uint_as_float(~e));
    AGG[(size_t)nbase * DIM + t] = v;
  }
}

// ---------------------------------------------------------------------------
// 2) GraphNet update: O = leaky([x | x_att[batch] | agg] @ Wa + ba) + x.
//    Three accumulated K=128 WMMA passes (Wa is [384,128] row-major).
//    grid.x = N/16, 8 waves cover the 128 output columns.
// ---------------------------------------------------------------------------
__global__ __launch_bounds__(256) void gemm_z_kernel(
    const float* __restrict__ X, const float* __restrict__ XATT,
    const float* __restrict__ AGG, const int* __restrict__ batch,
    const float* __restrict__ Wa, const float* __restrict__ ba,
    float* __restrict__ O) {
  __shared__ float Xt[16 * DIM];
  __shared__ float Gt[16 * DIM];
  __shared__ float At[16 * DIM];
  const int tid = threadIdx.x;
  const int row0 = blockIdx.x * 16;
  for (int t = tid; t < 16 * DIM; t += 256) {
    Xt[t] = X[(size_t)row0 * DIM + t];
    At[t] = AGG[(size_t)row0 * DIM + t];
    const int m = t >> 7, d = t & 127;
    Gt[t] = XATT[(size_t)batch[row0 + m] * DIM + d];
  }
  __syncthreads();

  const int wv = tid >> 5, lane = tid & 31;
  const int hi = lane >> 4, mn = lane & 15;
  const int col = wv * 16 + mn;

  v8f acc = {};
  for (int s = 0; s < 3; ++s) {
    const float* Ap = (s == 0) ? Xt : (s == 1) ? Gt : At;
    const float* Wp = Wa + (size_t)s * DIM * DIM;
    for (int k = 0; k < DIM; k += 4) {
      const int k2 = k + 2 * hi;
      acc = wmma_step(afrag(Ap, DIM, mn, k2), bfrag(Wp, DIM, k2, col), acc);
    }
  }
  const float bv = ba[col];
#pragma unroll
  for (int r = 0; r < 8; ++r) {
    const int m = r + 8 * hi;
    float v = leaky01(acc[r] + bv);
    v += Xt[m * DIM + col];  // residual
    O[(size_t)(row0 + m) * DIM + col] = v;
  }
}

// ---------------------------------------------------------------------------
// 3) Fully fused GATv2, one workgroup per graph, looping over heads:
//    WMMA sl/sr head projections into LDS, VALU logits, softmax,
//    WMMA alpha @ sl accumulated in registers across heads.
// ---------------------------------------------------------------------------
__global__ __launch_bounds__(256) void gat_fused_kernel(
    const float* __restrict__ X, const float* __restrict__ Wl,
    const float* __restrict__ bl, const float* __restrict__ Wr,
    const float* __restrict__ br, const float* __restrict__ attw,
    const float* __restrict__ attb, float* __restrict__ XATT) {
  __shared__ float xg[NPER * DIM];    // 32 KB
  __shared__ float sl[NPER * DIM];    // 32 KB
  __shared__ float sr[NPER * DIM];    // 32 KB
  __shared__ float lg[NPER * NPER];   // 16 KB logits -> alpha
  __shared__ float av[DIM];
  const int g = blockIdx.x, tid = threadIdx.x;
  for (int t = tid; t < NPER * DIM; t += 256)
    xg[t] = X[(size_t)(g * NPER) * DIM + t];

  const int wv = tid >> 5, lane = tid & 31;
  const int hi = lane >> 4, mn = lane & 15;
  const int rt = wv >> 1, cg = (wv & 1) * 4;

  v8f oacc[4] = {};  // per-wave 16x64 output block, summed over heads

  for (int h = 0; h < 3; ++h) {
    if (tid < DIM) av[tid] = attw[h * DIM + tid];
    __syncthreads();  // xg ready (h=0); prior head's reads of sl/lg done

    // sl = xg @ Wl[:, h*128 + .] + bl ; sr likewise (WMMA -> LDS)
    {
      v8f al[4] = {}, ar[4] = {};
      for (int k = 0; k < DIM; k += 4) {
        const int k2 = k + 2 * hi;
        const v2f a = afrag(xg, DIM, rt * 16 + mn, k2);
#pragma unroll
        for (int t = 0; t < 4; ++t) {
          const int col = h * DIM + (cg + t) * 16 + mn;
          al[t] = wmma_step(a, bfrag(Wl, 3 * DIM, k2, col), al[t]);
          ar[t] = wmma_step(a, bfrag(Wr, 3 * DIM, k2, col), ar[t]);
        }
      }
#pragma unroll
      for (int t = 0; t < 4; ++t) {
        const int c = (cg + t) * 16 + mn;
        const float bvl = bl[h * DIM + c], bvr = br[h * DIM + c];
#pragma unroll
        for (int r = 0; r < 8; ++r) {
          const int row = rt * 16 + r + 8 * hi;
          sl[row * DIM + c] = al[t][r] + bvl;
          sr[row * DIM + c] = ar[t][r] + bvr;
        }
      }
    }
    __syncthreads();

    // logits[i,j] = sum_d leaky0.2(sl[j,d] + sr[i,d]) * av[d]
    for (int p = tid; p < NPER * NPER; p += 256) {
      const int i = p >> 6, j = p & 63;
      float s = 0.f;
      for (int d = 0; d < DIM; ++d) {
        float v = sl[j * DIM + d] + sr[i * DIM + d];
        v = v < 0.f ? 0.2f * v : v;
        s = fmaf(v, av[d], s);
      }
      lg[p] = s;
    }
    __syncthreads();

    // softmax over incoming j per dst i (complete graph: never empty)
    if (tid < NPER) {
      const int i = tid;
      float mx = -3.4e38f;
      for (int j = 0; j < NPER; ++j) mx = fmaxf(mx, lg[i * NPER + j]);
      float sum = 0.f;
      for (int j = 0; j < NPER; ++j) {
        const float e = __expf(lg[i * NPER + j] - mx);
        lg[i * NPER + j] = e;
        sum += e;
      }
      const float inv = 1.f / sum;
      for (int j = 0; j < NPER; ++j) lg[i * NPER + j] *= inv;
    }
    __syncthreads();

    // oacc += alpha(64x64) @ sl(64x128)  (WMMA, K=64)
    for (int k = 0; k < NPER; k += 4) {
      const int k2 = k + 2 * hi;
      const v2f a = afrag(lg, NPER, rt * 16 + mn, k2);
#pragma unroll
      for (int t = 0; t < 4; ++t)
        oacc[t] = wmma_step(a, bfrag(sl, DIM, k2, (cg + t) * 16 + mn), oacc[t]);
    }
  }

  // x_att = leaky(mean_heads + att_bias)
#pragma unroll
  for (int t = 0; t < 4; ++t) {
    const int col = (cg + t) * 16 + mn;
    const float bv = attb[col];
#pragma unroll
    for (int r = 0; r < 8; ++r) {
      const int row = rt * 16 + r + 8 * hi;
      float v = oacc[t][r] * (1.f / 3.f) + bv;
      v = leaky01(v);
      XATT[(size_t)(g * NPER + row) * DIM + col] = v;
    }
  }
}

// ---------------------------------------------------------------------------
// Host orchestration. Input order (setup_inputs): x, edge_index, batch_ind,
// edge_complete, Wm, bm, Wa, ba, Wl, bl, Wr, br, att_w, att_bias.
// ---------------------------------------------------------------------------
extern "C" void kernel_launch(void* const* d_in, const int* in_sizes, int n_in,
                              void* d_out, int out_size, void* d_ws,
                              size_t ws_size, hipStream_t stream) {
  (void)in_sizes; (void)n_in; (void)out_size; (void)ws_size;
  const float* x0    = (const float*)d_in[0];
  const int*   eidx  = (const int*)d_in[1];
  const int*   batch = (const int*)d_in[2];
  // d_in[3] edge_complete: implicit (complete per-graph blocks)
  const float* Wm   = (const float*)d_in[4];
  const float* bm   = (const float*)d_in[5];
  const float* Wa   = (const float*)d_in[6];
  const float* ba   = (const float*)d_in[7];
  const float* Wl   = (const float*)d_in[8];
  const float* bl   = (const float*)d_in[9];
  const float* Wr   = (const float*)d_in[10];
  const float* br   = (const float*)d_in[11];
  const float* attw = (const float*)d_in[12];
  const float* attb = (const float*)d_in[13];
  float* out = (float*)d_out;

  float* ws   = (float*)d_ws;
  float* bufA = ws;
  float* bufB = bufA + (size_t)NNODES * DIM;
  float* xatt = bufB + (size_t)NNODES * DIM;
  float* agg  = xatt + (size_t)NNODES * DIM;
  // total workspace: 4 x 2 MB = 8 MB

  // step-0 x_att is zeros (graph-capturable async memset)
  hipMemsetAsync(xatt, 0, (size_t)NNODES * DIM * sizeof(float), stream);

  const float* xcur = x0;
  for (int i = 0; i < 4; ++i) {
    float* xout = (i == 3) ? out : ((i & 1) ? bufB : bufA);

    msg_segmax_kernel<<<NGRAPH, 256, 0, stream>>>(
        xcur, Wm + (size_t)i * DIM * DIM, bm + (size_t)i * DIM,
        eidx, eidx + NEDGE, agg);

    gemm_z_kernel<<<NNODES / 16, 256, 0, stream>>>(
        xcur, xatt, agg, batch, Wa + (size_t)i * 3 * DIM * DIM,
        ba + (size_t)i * DIM, xout);

    if (i < 3) {
      gat_fused_kernel<<<NGRAPH, 256, 0, stream>>>(
          xout, Wl + (size_t)i * DIM * 3 * DIM, bl + (size_t)i * 3 * DIM,
          Wr + (size_t)i * DIM * 3 * DIM, br + (size_t)i * 3 * DIM,
          attw + (size_t)i * 3 * DIM, attb + (size_t)i * DIM, xatt);
    }
    xcur = xout;
  }
}